// AdaptiveGaussianTrendCentered_63007170232744
// MI455X (gfx1250) — compile-verified
//
#include <hip/hip_runtime.h>
#include <math.h>

typedef float v2f __attribute__((ext_vector_type(2)));
typedef float v8f __attribute__((ext_vector_type(8)));

static constexpr int Bc = 16, Tc = 4096, Cc = 64, Sc = 5, Hc = 32;
static constexpr int TILE_T = 64;
static constexpr int HALO   = 56;                  // max radius
static constexpr int ROWS   = TILE_T + 2 * HALO;   // 176
static constexpr int XPITCH = 72;                  // halves hit disjoint bank sets

// Compile-time sigma tables (fold to immediates in unrolled loops)
static constexpr int   R_[Sc]     = {10, 16, 24, 36, 56};
static constexpr float SIG_[Sc]   = {2.5f, 4.0f, 6.0f, 9.0f, 14.0f};
// Zero-padded tap storage: per sigma [16 zeros][2R+1 taps][16 zeros]
static constexpr int   GBASE_[Sc] = {16, 69, 134, 215, 320};
static constexpr int   GKP_SIZE   = 449;

// Abramowitz-Stegun 7.1.26: |err| <= 1.5e-7 absolute, branchless.
__device__ __forceinline__ float fast_erff(float x) {
    float ax = fabsf(x);
    float d  = fmaf(0.3275911f, ax, 1.0f);
#if __has_builtin(__builtin_amdgcn_rcpf)
    float t  = __builtin_amdgcn_rcpf(d);
#else
    float t  = 1.0f / d;
#endif
    float p  = fmaf(t, 1.061405429f, -1.453152027f);
    p = fmaf(t, p, 1.421413741f);
    p = fmaf(t, p, -0.284496736f);
    p = fmaf(t, p, 0.254829592f);
    float e  = __expf(-ax * ax);
    float y  = fmaf(-p * t, e, 1.0f);
    return copysignf(y, x);
}

__global__ __launch_bounds__(256)
void agtc_kernel(const float* __restrict__ x,  const float* __restrict__ W1,
                 const float* __restrict__ b1, const float* __restrict__ W2,
                 const float* __restrict__ b2, float* __restrict__ out)
{
    __shared__ __align__(16) float Xs[ROWS * XPITCH];   // ~49.5 KB
    __shared__ float gkp[GKP_SIZE];                      // zero-padded taps
    __shared__ float w1s[64];
    __shared__ float b1s[32];
    __shared__ float w2s[160];
    __shared__ float b2s[8];

    const int tid = threadIdx.x;
    const int t0  = blockIdx.x * TILE_T;
    const int bb  = blockIdx.y;

    // ---- zero padded tap table; stage MLP weights; stage x tile + halo ----
    for (int i = tid; i < GKP_SIZE; i += 256) gkp[i] = 0.0f;
    if (tid < 64)  w1s[tid] = W1[tid];
    if (tid < 32)  b1s[tid] = b1[tid];
    if (tid < 160) w2s[tid] = W2[tid];
    if (tid < 5)   b2s[tid] = b2[tid];

    const float* xb = x + (size_t)bb * Tc * Cc;
    for (int idx = tid; idx < ROWS * 16; idx += 256) {   // 176 rows x 16 float4
        int row = idx >> 4;
        int c4  = (idx & 15) << 2;
        int g = t0 - HALO + row;                          // reflect (jnp.pad 'reflect')
        if (g < 0)      g = -g;
        if (g > Tc - 1) g = 2 * (Tc - 1) - g;
        float4 v = *reinterpret_cast<const float4*>(xb + (size_t)g * Cc + c4);
        *reinterpret_cast<float4*>(&Xs[row * XPITCH + c4]) = v;
    }
    __syncthreads();

    // ---- fill normalized Gaussian taps (thread s handles sigma s) ----
#pragma unroll
    for (int s = 0; s < Sc; ++s) {
        if (tid == s) {
            const int   R    = R_[s];
            const int   base = GBASE_[s];
            const float is   = 1.0f / SIG_[s];
            float sum = 0.0f;
            for (int j = 0; j <= 2 * R; ++j) {
                float u = (float)(j - R) * is;
                float v = expf(-0.5f * u * u);
                gkp[base + j] = v;
                sum += v;
            }
            float inv = 1.0f / (sum + 1e-12f);
            for (int j = 0; j <= 2 * R; ++j) gkp[base + j] *= inv;
        }
    }
    __syncthreads();

    const int wave = tid >> 5;
    const int lane = tid & 31;
    const int half = lane >> 4;     // K = vgpr + 2*half (A,B); M = vgpr + 8*half (C/D)
    const int ln   = lane & 15;

    // Wave owns two c-tiles of the SAME t sub-tile: tiles 2w (cc0) and 2w+1 (cc1).
    const int tt  = wave >> 1;
    const int cb0 = ((2 * wave) & 3) << 4;
    const int cb1 = cb0 + 16;

    v8f acc[2][Sc];
#pragma unroll
    for (int j2 = 0; j2 < 2; ++j2)
#pragma unroll
        for (int s = 0; s < Sc; ++s) acc[j2][s] = {};

    // ---- banded-matmul conv: fully unrolled, branchless, shared A fragment ----
#pragma unroll
    for (int s = 0; s < Sc; ++s) {
        const int R       = R_[s];
        const int base    = GBASE_[s];
        const int span    = 16 + 2 * R;            // 36/48/64/88/128
        const int rowbase = 16 * tt - R + HALO;
#pragma unroll
        for (int k0 = 0; k0 < span; k0 += 4) {
            v2f a, f0, f1;
#pragma unroll
            for (int v = 0; v < 2; ++v) {
                const int k = k0 + v + 2 * half;
                a[v]  = gkp[base + k - ln];        // zero pads handle out-of-band
                f0[v] = Xs[(rowbase + k) * XPITCH + cb0 + ln];
                f1[v] = Xs[(rowbase + k) * XPITCH + cb1 + ln];
            }
            acc[0][s] = __builtin_amdgcn_wmma_f32_16x16x4_f32(
                            false, a, false, f0, (short)0, acc[0][s], false, false);
            acc[1][s] = __builtin_amdgcn_wmma_f32_16x16x4_f32(
                            false, a, false, f1, (short)0, acc[1][s], false, false);
        }
    }

    // ---- per c-tile: rolling causal stats -> hidden-major MLP -> softmax ----
    constexpr float INV_TEMP = 1.0f / 0.7f;
    float* ob = out + (size_t)bb * Tc * Cc;
    const int tb = t0 + 16 * tt + 8 * half;        // lane's first t (r = 0)

#pragma unroll
    for (int j2 = 0; j2 < 2; ++j2) {
        const int cch = (j2 ? cb1 : cb0) + ln;

        // rolling window sums (left replicate padding via clamp); exact:
        // S(t+1) = S(t) + x[t+1] - x[max(t-15,0)]
        float z8[8], lv8[8];
        {
            float sx = 0.0f, sx2 = 0.0f, xcur = 0.0f;
#pragma unroll
            for (int dt = 0; dt < 16; ++dt) {
                int g = tb - dt; if (g < 0) g = 0;
                float xv = Xs[(g - t0 + HALO) * XPITCH + cch];
                sx += xv; sx2 += xv * xv;
                if (dt == 0) xcur = xv;
            }
#pragma unroll
            for (int r = 0; r < 8; ++r) {
                const int t = tb + r;
                const float eff  = fminf((float)(t + 1), 16.0f) + 1e-12f;
                const float ie   = 1.0f / eff;
                const float mean = sx * ie;
                const float var  = fmaxf(sx2 * ie - mean * mean, 0.0f);
                z8[r]  = (xcur - mean) / sqrtf(var + 1e-6f);
                lv8[r] = __logf(var + 1e-6f);
                if (r < 7) {
                    float xnew = Xs[(t + 1 - t0 + HALO) * XPITCH + cch];
                    int g = t - 15; if (g < 0) g = 0;
                    float xold = Xs[(g - t0 + HALO) * XPITCH + cch];
                    sx  += xnew - xold;
                    sx2 += xnew * xnew - xold * xold;
                    xcur = xnew;
                }
            }
        }

        // hidden-major MLP: weights loaded once per j, reused by all 8 elements
        float lg[8][Sc];
#pragma unroll
        for (int r = 0; r < 8; ++r)
#pragma unroll
            for (int s = 0; s < Sc; ++s) lg[r][s] = b2s[s];

        for (int j = 0; j < Hc; ++j) {
            const float w10 = w1s[2 * j];
            const float w11 = w1s[2 * j + 1];
            const float bj  = b1s[j];
            float w2j[Sc];
#pragma unroll
            for (int s = 0; s < Sc; ++s) w2j[s] = w2s[s * Hc + j];
#pragma unroll
            for (int r = 0; r < 8; ++r) {
                float pre = fmaf(z8[r], w10, fmaf(lv8[r], w11, bj));
                float hj  = 0.5f * pre * (1.0f + fast_erff(pre * 0.70710678118654752f));
#pragma unroll
                for (int s = 0; s < Sc; ++s) lg[r][s] = fmaf(hj, w2j[s], lg[r][s]);
            }
        }

        // temperature softmax + weighted combine of the 5 WMMA accumulators
#pragma unroll
        for (int r = 0; r < 8; ++r) {
            float mx = lg[r][0];
#pragma unroll
            for (int s = 1; s < Sc; ++s) mx = fmaxf(mx, lg[r][s]);
            float se = 0.0f, num = 0.0f;
#pragma unroll
            for (int s = 0; s < Sc; ++s) {
                float e = __expf((lg[r][s] - mx) * INV_TEMP);
                se += e;
                num = fmaf(acc[j2][s][r], e, num);
            }
            ob[(size_t)(tb + r) * Cc + cch] = num / se;
        }
    }
}

extern "C" void kernel_launch(void* const* d_in, const int* in_sizes, int n_in,
                              void* d_out, int out_size, void* d_ws, size_t ws_size,
                              hipStream_t stream) {
    (void)in_sizes; (void)n_in; (void)out_size; (void)d_ws; (void)ws_size;
    const float* x  = (const float*)d_in[0];
    const float* W1 = (const float*)d_in[1];
    const float* b1 = (const float*)d_in[2];
    const float* W2 = (const float*)d_in[3];
    const float* b2 = (const float*)d_in[4];
    float* out = (float*)d_out;

    dim3 grid(Tc / TILE_T, Bc);   // (64, 16)
    agtc_kernel<<<grid, 256, 0, stream>>>(x, W1, b1, W2, b2, out);
}